// CapsLayer_19172734009880
// MI455X (gfx1250) — compile-verified
//
#include <hip/hip_runtime.h>

// Problem constants (match reference)
#define B_    32
#define IC_   1152
#define OC_   10
#define ID_   8
#define OD_   16
#define NIT   5
#define EPSF  1e-20f

#define NBLK  288                  // workgroups in main kernel
#define IPB   (IC_ / NBLK)         // 4 capsules per block
#define TPB   (B_ * OC_)           // 320 threads = 10 waves (wave==oc, lane==b)
#define WELT  (OC_ * ID_ * OD_)    // 1280 floats per capsule weight slab (5 KB)
#define OUTN  (B_ * OC_ * OD_)     // 5120 outputs

typedef __attribute__((ext_vector_type(2))) float v2f;
typedef __attribute__((ext_vector_type(8))) float v8f;

__device__ __forceinline__ float rcp_fast(float x) {
    return __builtin_amdgcn_rcpf(x);   // v_rcp_f32 (trans op), plenty for 1e-20-guarded divides
}

// CDNA5 async global->LDS copy (ASYNCcnt-tracked). 16 bytes per lane.
__device__ __forceinline__ void async_ld_b128(uint32_t lds_byte_addr, const void* gsrc) {
    asm volatile("global_load_async_to_lds_b128 %0, %1, off"
                 :: "v"(lds_byte_addr), "v"(gsrc)
                 : "memory");
}
__device__ __forceinline__ void wait_async0() {
    asm volatile("s_wait_asynccnt 0" ::: "memory");
}

__global__ __launch_bounds__(TPB) void caps_nnmf_kernel(
    const float* __restrict__ x,        // [B, IC, ID]
    const float* __restrict__ w,        // [IC, OC, ID, OD]
    float* __restrict__ partial)        // [NBLK, B, OC, OD]
{
    // Double-buffered weight slab + alpha exchange buffer (11.5 KB of 320 KB LDS)
    __shared__ float4 shw[2][WELT / 4];
    __shared__ float  sh_alpha[TPB];

    const int tid = threadIdx.x;
    const int b   = tid & 31;          // lane
    const int oc  = tid >> 5;          // wave index 0..9

    const uint32_t lds0 = (uint32_t)(uintptr_t)(&shw[0][0]);
    const uint32_t lds1 = (uint32_t)(uintptr_t)(&shw[1][0]);

    const int i0 = blockIdx.x * IPB;

    // Stage first slab: 320 lanes x 16B = 5120B = whole w[i0]
    async_ld_b128(lds0 + (uint32_t)tid * 16u,
                  (const char*)(w + (size_t)i0 * WELT) + (size_t)tid * 16);
    wait_async0();
    __syncthreads();

    float acc[OD_];
#pragma unroll
    for (int od = 0; od < OD_; ++od) acc[od] = 0.f;

    int cur = 0;
    for (int ii = 0; ii < IPB; ++ii) {
        const int i = i0 + ii;

        // Prefetch next slab into the other buffer (overlaps with compute below)
        if (ii + 1 < IPB) {
            async_ld_b128((cur ? lds0 : lds1) + (uint32_t)tid * 16u,
                          (const char*)(w + (size_t)(i + 1) * WELT) + (size_t)tid * 16);
        }

        // Load and normalize x[b, i, :] over ID
        float xn[ID_];
        {
            const float* xp = x + ((size_t)b * IC_ + i) * ID_;
            float s = 0.f;
#pragma unroll
            for (int id = 0; id < ID_; ++id) { xn[id] = xp[id]; s += xn[id]; }
            float r = rcp_fast(s + EPSF);
#pragma unroll
            for (int id = 0; id < ID_; ++id) xn[id] *= r;
        }

        // This wave's 8x16 slice of w[i]: 32 float4, wave-uniform address (broadcast reads)
        const float4* wv = &shw[cur][(size_t)oc * (ID_ * OD_ / 4)];

        float o0=1.f/OD_,o1=1.f/OD_,o2=1.f/OD_,o3=1.f/OD_,o4=1.f/OD_,o5=1.f/OD_,o6=1.f/OD_,o7=1.f/OD_,
              o8=1.f/OD_,o9=1.f/OD_,oa=1.f/OD_,ob=1.f/OD_,oc_=1.f/OD_,od_=1.f/OD_,oe=1.f/OD_,of_=1.f/OD_;

#pragma unroll
        for (int it = 0; it < NIT; ++it) {
            float s0=0,s1=0,s2=0,s3=0,s4=0,s5=0,s6=0,s7=0,
                  s8=0,s9=0,sa=0,sb=0,sc=0,sd=0,se=0,sf=0;
#pragma unroll
            for (int id = 0; id < ID_; ++id) {
                float4 w0 = wv[id*4+0], w1 = wv[id*4+1], w2 = wv[id*4+2], w3 = wv[id*4+3];
                // denom[id] = <w_row, out>
                float d = w0.x*o0 + w0.y*o1 + w0.z*o2 + w0.w*o3
                        + w1.x*o4 + w1.y*o5 + w1.z*o6 + w1.w*o7
                        + w2.x*o8 + w2.y*o9 + w2.z*oa + w2.w*ob
                        + w3.x*oc_ + w3.y*od_ + w3.z*oe + w3.w*of_;
                float cid = xn[id] * rcp_fast(d + EPSF);
                s0 += cid*w0.x; s1 += cid*w0.y; s2 += cid*w0.z; s3 += cid*w0.w;
                s4 += cid*w1.x; s5 += cid*w1.y; s6 += cid*w1.z; s7 += cid*w1.w;
                s8 += cid*w2.x; s9 += cid*w2.y; sa += cid*w2.z; sb += cid*w2.w;
                sc += cid*w3.x; sd += cid*w3.y; se += cid*w3.z; sf += cid*w3.w;
            }
            // out_new = (out*s) / (sum(out*s) + eps)
            o0*=s0; o1*=s1; o2*=s2; o3*=s3; o4*=s4; o5*=s5; o6*=s6; o7*=s7;
            o8*=s8; o9*=s9; oa*=sa; ob*=sb; oc_*=sc; od_*=sd; oe*=se; of_*=sf;
            float tsum = ((o0+o1)+(o2+o3)) + ((o4+o5)+(o6+o7))
                       + ((o8+o9)+(oa+ob)) + ((oc_+od_)+(oe+of_));
            float r = rcp_fast(tsum + EPSF);
            o0*=r; o1*=r; o2*=r; o3*=r; o4*=r; o5*=r; o6*=r; o7*=r;
            o8*=r; o9*=r; oa*=r; ob*=r; oc_*=r; od_*=r; oe*=r; of_*=r;
        }

        // reconstruct[id] = <h, w_row>;  alpha_raw = sum_id xn[id]*reconstruct[id]
        float araw = 0.f;
#pragma unroll
        for (int id = 0; id < ID_; ++id) {
            float4 w0 = wv[id*4+0], w1 = wv[id*4+1], w2 = wv[id*4+2], w3 = wv[id*4+3];
            float rec = w0.x*o0 + w0.y*o1 + w0.z*o2 + w0.w*o3
                      + w1.x*o4 + w1.y*o5 + w1.z*o6 + w1.w*o7
                      + w2.x*o8 + w2.y*o9 + w2.z*oa + w2.w*ob
                      + w3.x*oc_ + w3.y*od_ + w3.z*oe + w3.w*of_;
            araw += xn[id] * rec;
        }

        // alpha normalization across the 10 oc's of this (b,i)
        sh_alpha[tid] = araw;          // [oc][b]
        __syncthreads();
        float asum = 0.f;
#pragma unroll
        for (int o2i = 0; o2i < OC_; ++o2i) asum += sh_alpha[o2i * 32 + b];
        float alpha = araw * rcp_fast(asum + EPSF);

        acc[0]+=alpha*o0;  acc[1]+=alpha*o1;  acc[2]+=alpha*o2;  acc[3]+=alpha*o3;
        acc[4]+=alpha*o4;  acc[5]+=alpha*o5;  acc[6]+=alpha*o6;  acc[7]+=alpha*o7;
        acc[8]+=alpha*o8;  acc[9]+=alpha*o9;  acc[10]+=alpha*oa; acc[11]+=alpha*ob;
        acc[12]+=alpha*oc_;acc[13]+=alpha*od_;acc[14]+=alpha*oe; acc[15]+=alpha*of_;

        // Next-slab data must be resident; also guards sh_alpha reuse next ii
        wait_async0();
        __syncthreads();
        cur ^= 1;
    }

    // partial[blk][b][oc][od]
    float4* pp = (float4*)partial + (((size_t)blockIdx.x * B_ + b) * OC_ + oc) * 4;
    pp[0] = make_float4(acc[0],  acc[1],  acc[2],  acc[3]);
    pp[1] = make_float4(acc[4],  acc[5],  acc[6],  acc[7]);
    pp[2] = make_float4(acc[8],  acc[9],  acc[10], acc[11]);
    pp[3] = make_float4(acc[12], acc[13], acc[14], acc[15]);
}

// ---------------------------------------------------------------------------
// Deterministic cross-block reduction on the matrix pipe:
//   out[j] = sum_{k<NBLK} partial[k][j]  ==  D = A x Ones + C  with
//   V_WMMA_F32_16X16X4_F32.  Each wave reduces 16 outputs (A's M rows),
//   K-chunks of 4 partials per WMMA, 4 round-robin accumulators to break the
//   C->C dependency chain.  Fixed accumulation order -> bitwise deterministic.
// ---------------------------------------------------------------------------
#define RED_WPB    8                         // waves per block
#define RED_TPB    (RED_WPB * 32)
#define RED_BLOCKS (OUTN / 16 / RED_WPB)     // 5120/16 = 320 waves -> 40 blocks

__device__ __forceinline__ v2f red_load_a(const float* pbase, int kk, int khalf) {
    v2f a;
    a.x = pbase[(size_t)(kk + khalf)     * OUTN];   // A vgpr0: K = kk   (+2 for hi lanes)
    a.y = pbase[(size_t)(kk + khalf + 1) * OUTN];   // A vgpr1: K = kk+1 (+2 for hi lanes)
    return a;
}

__global__ __launch_bounds__(RED_TPB) void caps_reduce_wmma_kernel(
    const float* __restrict__ partial, float* __restrict__ out)
{
    const int lane  = threadIdx.x & 31;
    const int wave  = threadIdx.x >> 5;
    const int jbase = (blockIdx.x * RED_WPB + wave) * 16;

    const int mrow  = lane & 15;            // A layout: M = lane%16
    const int khalf = (lane >> 4) << 1;     // lanes 16-31 hold K+2 / K+3

    const float* pbase = partial + jbase + mrow;

    v2f ones; ones.x = 1.f; ones.y = 1.f;   // B = 4x16 all-ones
    v8f c0 = {}, c1 = {}, c2 = {}, c3 = {};

    // NBLK = 288 partials: 72 WMMAs of K=4, interleaved over 4 accumulators
    for (int kk = 0; kk < NBLK; kk += 16) {
        v2f a0 = red_load_a(pbase, kk + 0,  khalf);
        v2f a1 = red_load_a(pbase, kk + 4,  khalf);
        v2f a2 = red_load_a(pbase, kk + 8,  khalf);
        v2f a3 = red_load_a(pbase, kk + 12, khalf);
        c0 = __builtin_amdgcn_wmma_f32_16x16x4_f32(false, a0, false, ones, (short)0, c0, false, false);
        c1 = __builtin_amdgcn_wmma_f32_16x16x4_f32(false, a1, false, ones, (short)0, c1, false, false);
        c2 = __builtin_amdgcn_wmma_f32_16x16x4_f32(false, a2, false, ones, (short)0, c2, false, false);
        c3 = __builtin_amdgcn_wmma_f32_16x16x4_f32(false, a3, false, ones, (short)0, c3, false, false);
    }
    v8f c = (c0 + c1) + (c2 + c3);

    // C/D layout: VGPR r on lane l holds D[M = r + 8*(l/16)][N = l%16]; every N
    // column is identical (B==1), so lanes 0 and 16 each store 8 results.
    if (lane == 0 || lane == 16) {
        float* op = out + jbase + ((lane >> 4) << 3);
#pragma unroll
        for (int r = 0; r < 8; ++r) op[r] = c[r];
    }
}

extern "C" void kernel_launch(void* const* d_in, const int* in_sizes, int n_in,
                              void* d_out, int out_size, void* d_ws, size_t ws_size,
                              hipStream_t stream) {
    const float* x = (const float*)d_in[0];   // [32,1152,8]
    const float* w = (const float*)d_in[1];   // [1152,10,8,16]
    float* partial = (float*)d_ws;            // needs NBLK*OUTN*4 = ~5.9 MB
    caps_nnmf_kernel<<<NBLK, TPB, 0, stream>>>(x, w, partial);
    caps_reduce_wmma_kernel<<<RED_BLOCKS, RED_TPB, 0, stream>>>(partial, (float*)d_out);
}